// PillarEncoder_87454124081813
// MI455X (gfx1250) — compile-verified
//
#include <hip/hip_runtime.h>
#include <hip/hip_bf16.h>

typedef float v2f __attribute__((ext_vector_type(2)));
typedef float v8f __attribute__((ext_vector_type(8)));

#define NB        256          // blocks for stat/final passes
#define WAVES     2
#define BLOCK     (WAVES * 32)
#define N_PILLARS 40000
#define PPT       32
#define BN_CNT    1280000.0f   // N * P
#define BN_EPS    0.001f

// ---- workspace layout (float offsets) ----
#define OFF_SCALE0 0
#define OFF_SHIFT0 32
#define OFF_SCALE1 64
#define OFF_SHIFT1 192
#define OFF_PSUM0  320
#define OFF_PSQ0   (OFF_PSUM0 + NB * 32)
#define OFF_PSUM1  (OFF_PSQ0 + NB * 32)
#define OFF_PSQ1   (OFF_PSUM1 + NB * 128)

struct SMem {
  float w0s[12 * 32];          // w0 transposed + K-padded: [k][o]
  float w1s[64 * 128];         // w1 transposed: [k][o]
  float f0s[WAVES][32 * 12];   // per-wave feats (32 rows x 11 + pad)
  float f1s[WAVES][32 * 64];   // per-wave layer-1 input
  float red[WAVES][256];       // cross-wave deterministic stat reduction
};

__device__ __forceinline__ v8f wmma4(v2f a, v2f b, v8f c) {
  // D(16x16,f32) = A(16x4,f32) x B(4x16,f32) + C   -> v_wmma_f32_16x16x4_f32
  return __builtin_amdgcn_wmma_f32_16x16x4_f32(false, a, false, b, (short)0, c,
                                               false, false);
}

// MODE 0: layer-0 raw GEMM, accumulate per-channel sum/sumsq of y0
// MODE 1: bn0+relu+max -> feats1 -> layer-1 raw GEMM, accumulate y1 stats
// MODE 2: full recompute, bn1+relu+max_P -> out (N,128)
template <int MODE>
__global__ __launch_bounds__(BLOCK) void pfn_pass(
    const float* __restrict__ vf, const int* __restrict__ coords,
    const int* __restrict__ vnp, const float* __restrict__ w0,
    const float* __restrict__ w1, float* __restrict__ ws,
    float* __restrict__ out) {
  __shared__ SMem sm;
  const int tid  = threadIdx.x;
  const int wave = tid >> 5;
  const int lane = tid & 31;
  const int h    = lane >> 4;   // lane half-group (0/1)
  const int ln   = lane & 15;

  // stage w0 transposed [k][o], pad K 11->12 with zeros
  for (int i = tid; i < 12 * 32; i += BLOCK) {
    const int k = i >> 5, o = i & 31;
    sm.w0s[i] = (k < 11) ? w0[o * 11 + k] : 0.0f;
  }
  if (MODE >= 1) {
    for (int i = tid; i < 128 * 64; i += BLOCK) {
      const int o = i >> 6, c = i & 63;
      sm.w1s[c * 128 + o] = w1[i];
    }
  }
  __syncthreads();

  float sc0[2], sh0[2];
  if (MODE >= 1) {
#pragma unroll
    for (int t = 0; t < 2; ++t) {
      sc0[t] = ws[OFF_SCALE0 + t * 16 + ln];
      sh0[t] = ws[OFF_SHIFT0 + t * 16 + ln];
    }
  }
  float sc1[8], sh1[8];
  if (MODE == 2) {
#pragma unroll
    for (int t = 0; t < 8; ++t) {
      sc1[t] = ws[OFF_SCALE1 + t * 16 + ln];
      sh1[t] = ws[OFF_SHIFT1 + t * 16 + ln];
    }
  }

  float s0[2] = {0.f, 0.f}, q0[2] = {0.f, 0.f};
  float s1[8] = {0.f, 0.f, 0.f, 0.f, 0.f, 0.f, 0.f, 0.f};
  float q1[8] = {0.f, 0.f, 0.f, 0.f, 0.f, 0.f, 0.f, 0.f};

  const int waveStride = gridDim.x * WAVES;
  for (int n = blockIdx.x * WAVES + wave; n < N_PILLARS; n += waveStride) {
    // ---------------- build 11 masked features (lane = point p) -----------
    const float4 p = reinterpret_cast<const float4*>(vf)[n * PPT + lane];
    const float x = p.x, y = p.y, z = p.z, inten = p.w;
    float sx = x, sy = y, sz = z;
#pragma unroll
    for (int m = 16; m >= 1; m >>= 1) {
      sx += __shfl_xor(sx, m);
      sy += __shfl_xor(sy, m);
      sz += __shfl_xor(sz, m);
    }
    const int np    = vnp[n];
    const float inv = 1.0f / (float)np;
    const float mx = sx * inv, my = sy * inv, mz = sz * inv;
    const float dist = sqrtf(x * x + y * y + z * z);
    const int c1 = coords[n * 4 + 1], c2 = coords[n * 4 + 2],
              c3 = coords[n * 4 + 3];
    const float cmx = (float)c3 * 0.16f + 0.08f;
    const float cmy = (float)c2 * 0.16f + (0.08f - 39.68f);
    const float cmz = (float)c1 * 4.0f + (2.0f - 3.0f);
    const float mk  = (lane < np) ? 1.0f : 0.0f;
    float* f0 = &sm.f0s[wave][lane * 12];
    f0[0] = x * mk;          f0[1] = y * mk;          f0[2] = z * mk;
    f0[3] = dist * mk;       f0[4] = inten * mk;
    f0[5] = (x - mx) * mk;   f0[6] = (y - my) * mk;   f0[7] = (z - mz) * mk;
    f0[8] = (x - cmx) * mk;  f0[9] = (y - cmy) * mk;  f0[10] = (z - cmz) * mk;
    f0[11] = 0.0f;
    __builtin_amdgcn_wave_barrier();

    // ---------------- layer 0: y0(32x32) = feats(32x12) @ w0t -------------
    float ymax[2] = {0.f, 0.f};  // relu output is >= 0
#pragma unroll
    for (int m = 0; m < 2; ++m) {
      const float* fr = &sm.f0s[wave][(m * 16 + ln) * 12];
#pragma unroll
      for (int t = 0; t < 2; ++t) {
        v8f acc = {};
#pragma unroll
        for (int s = 0; s < 3; ++s) {
          v2f a, b;
          a.x = fr[4 * s + 2 * h];
          a.y = fr[4 * s + 2 * h + 1];
          b.x = sm.w0s[(4 * s + 2 * h) * 32 + t * 16 + ln];
          b.y = sm.w0s[(4 * s + 2 * h + 1) * 32 + t * 16 + ln];
          acc = wmma4(a, b, acc);
        }
        if (MODE == 0) {
#pragma unroll
          for (int v = 0; v < 8; ++v) {
            s0[t] += acc[v];
            q0[t] += acc[v] * acc[v];
          }
        } else {
#pragma unroll
          for (int v = 0; v < 8; ++v) {
            const float yv = fmaxf(acc[v] * sc0[t] + sh0[t], 0.0f);
            sm.f1s[wave][(m * 16 + v + 8 * h) * 64 + t * 16 + ln] = yv;
            ymax[t] = fmaxf(ymax[t], yv);
          }
        }
      }
    }

    if (MODE >= 1) {
#pragma unroll
      for (int t = 0; t < 2; ++t)
        ymax[t] = fmaxf(ymax[t], __shfl_xor(ymax[t], 16));
      // broadcast y_max into columns 32..63; halves cover rows 0..31
#pragma unroll
      for (int t = 0; t < 2; ++t)
        for (int r = 0; r < 16; ++r)
          sm.f1s[wave][(h * 16 + r) * 64 + 32 + t * 16 + ln] = ymax[t];
      __builtin_amdgcn_wave_barrier();

      // -------------- layer 1: y1(32x128) = feats1(32x64) @ w1t -----------
      float omax[8] = {0.f, 0.f, 0.f, 0.f, 0.f, 0.f, 0.f, 0.f};
#pragma unroll
      for (int m = 0; m < 2; ++m) {
        const float* fr = &sm.f1s[wave][(m * 16 + ln) * 64];
#pragma unroll
        for (int t = 0; t < 8; ++t) {
          v8f acc = {};
#pragma unroll
          for (int s = 0; s < 16; ++s) {
            v2f a, b;
            a.x = fr[4 * s + 2 * h];
            a.y = fr[4 * s + 2 * h + 1];
            b.x = sm.w1s[(4 * s + 2 * h) * 128 + t * 16 + ln];
            b.y = sm.w1s[(4 * s + 2 * h + 1) * 128 + t * 16 + ln];
            acc = wmma4(a, b, acc);
          }
          if (MODE == 1) {
#pragma unroll
            for (int v = 0; v < 8; ++v) {
              s1[t] += acc[v];
              q1[t] += acc[v] * acc[v];
            }
          } else {
#pragma unroll
            for (int v = 0; v < 8; ++v)
              omax[t] = fmaxf(omax[t], fmaxf(acc[v] * sc1[t] + sh1[t], 0.0f));
          }
        }
      }
      if (MODE == 2) {
#pragma unroll
        for (int t = 0; t < 8; ++t)
          omax[t] = fmaxf(omax[t], __shfl_xor(omax[t], 16));
        if (h == 0) {
#pragma unroll
          for (int t = 0; t < 8; ++t)
            out[n * 128 + t * 16 + ln] = omax[t];
        }
      }
      __builtin_amdgcn_wave_barrier();
    }
  }

  // ---------------- deterministic per-block stat partials ------------------
  if (MODE == 0) {
#pragma unroll
    for (int t = 0; t < 2; ++t) {
      s0[t] += __shfl_xor(s0[t], 16);
      q0[t] += __shfl_xor(q0[t], 16);
    }
    if (h == 0) {
#pragma unroll
      for (int t = 0; t < 2; ++t) {
        sm.red[wave][t * 16 + ln]      = s0[t];
        sm.red[wave][32 + t * 16 + ln] = q0[t];
      }
    }
    __syncthreads();
    if (tid < 32) {
      float S = 0.f, Q = 0.f;
      for (int w = 0; w < WAVES; ++w) {
        S += sm.red[w][tid];
        Q += sm.red[w][32 + tid];
      }
      ws[OFF_PSUM0 + blockIdx.x * 32 + tid] = S;
      ws[OFF_PSQ0 + blockIdx.x * 32 + tid]  = Q;
    }
  } else if (MODE == 1) {
#pragma unroll
    for (int t = 0; t < 8; ++t) {
      s1[t] += __shfl_xor(s1[t], 16);
      q1[t] += __shfl_xor(q1[t], 16);
    }
    if (h == 0) {
#pragma unroll
      for (int t = 0; t < 8; ++t) {
        sm.red[wave][t * 16 + ln]       = s1[t];
        sm.red[wave][128 + t * 16 + ln] = q1[t];
      }
    }
    __syncthreads();
    for (int c = tid; c < 128; c += BLOCK) {
      float S = 0.f, Q = 0.f;
      for (int w = 0; w < WAVES; ++w) {
        S += sm.red[w][c];
        Q += sm.red[w][128 + c];
      }
      ws[OFF_PSUM1 + blockIdx.x * 128 + c] = S;
      ws[OFF_PSQ1 + blockIdx.x * 128 + c]  = Q;
    }
  }
}

// Fixed-order reduction of block partials -> scale/shift (deterministic).
__global__ void bn_finalize(const float* __restrict__ g,
                            const float* __restrict__ b, float* __restrict__ ws,
                            int nch, int psum, int psq, int oscale,
                            int oshift) {
  const int c = threadIdx.x;
  if (c >= nch) return;
  float S = 0.f, Q = 0.f;
  for (int i = 0; i < NB; ++i) {
    S += ws[psum + i * nch + c];
    Q += ws[psq + i * nch + c];
  }
  const float mean = S / BN_CNT;
  const float var  = Q / BN_CNT - mean * mean;
  const float rstd = rsqrtf(var + BN_EPS);
  const float sc   = g[c] * rstd;
  ws[oscale + c] = sc;
  ws[oshift + c] = b[c] - mean * sc;
}

extern "C" void kernel_launch(void* const* d_in, const int* in_sizes, int n_in,
                              void* d_out, int out_size, void* d_ws,
                              size_t ws_size, hipStream_t stream) {
  (void)in_sizes; (void)n_in; (void)out_size; (void)ws_size;
  const float* vf     = (const float*)d_in[0];
  const int* coords   = (const int*)d_in[1];
  const int* vnp      = (const int*)d_in[2];
  const float* w0     = (const float*)d_in[3];
  const float* g0     = (const float*)d_in[4];
  const float* b0     = (const float*)d_in[5];
  const float* w1     = (const float*)d_in[6];
  const float* g1     = (const float*)d_in[7];
  const float* b1     = (const float*)d_in[8];
  float* ws  = (float*)d_ws;
  float* out = (float*)d_out;

  pfn_pass<0><<<NB, BLOCK, 0, stream>>>(vf, coords, vnp, w0, w1, ws, out);
  bn_finalize<<<1, 32, 0, stream>>>(g0, b0, ws, 32, OFF_PSUM0, OFF_PSQ0,
                                    OFF_SCALE0, OFF_SHIFT0);
  pfn_pass<1><<<NB, BLOCK, 0, stream>>>(vf, coords, vnp, w0, w1, ws, out);
  bn_finalize<<<1, 128, 0, stream>>>(g1, b1, ws, 128, OFF_PSUM1, OFF_PSQ1,
                                     OFF_SCALE1, OFF_SHIFT1);
  pfn_pass<2><<<NB, BLOCK, 0, stream>>>(vf, coords, vnp, w0, w1, ws, out);
}